// aggg_GCN3_16226386444394
// MI455X (gfx1250) — compile-verified
//
#include <hip/hip_runtime.h>
#include <hip/hip_bf16.h>

// ---------------- problem constants (match reference) ----------------
#define NN      50000       // nodes
#define NE      1600000     // edges
#define HD      128         // F_IN == H == 128
#define NC      10          // classes
#define LD_REP  384         // 3*H
#define EPSV    1e-12f

typedef __attribute__((ext_vector_type(2))) float v2f;
typedef __attribute__((ext_vector_type(8))) float v8f;

// ---------------- degree / dinv ----------------
__global__ void k_deg_init(float* __restrict__ deg) {
    int i = blockIdx.x * blockDim.x + threadIdx.x;
    if (i < NN) deg[i] = 1.0f;              // self-loop weight
}

__global__ void k_deg_acc(const int* __restrict__ dst,
                          const float* __restrict__ w,
                          float* __restrict__ deg) {
    int e = blockIdx.x * blockDim.x + threadIdx.x;
    if (e < NE) atomicAdd(&deg[dst[e]], w[e]);
}

__global__ void k_dinv(float* __restrict__ deg_dinv) {
    int i = blockIdx.x * blockDim.x + threadIdx.x;
    if (i < NN) deg_dinv[i] = rsqrtf(deg_dinv[i]);   // deg >= 1 always
}

// ---------------- fp32 WMMA GEMM: h = X(ldx) @ W[128x128]; agg = b + dinv^2*h ----------------
// One wave computes a 16(M) x 128(N) strip. 3125 strips total.
__global__ void k_gemm_wmma(const float* __restrict__ X, int ldx,
                            const float* __restrict__ W,
                            const float* __restrict__ b,
                            const float* __restrict__ dinv,
                            float* __restrict__ h,
                            float* __restrict__ agg) {
    const int wave = blockIdx.x * (blockDim.x >> 5) + (threadIdx.x >> 5);
    const int lane = threadIdx.x & 31;
    if (wave >= NN / 16) return;

    const int m0    = wave * 16;
    const int l16   = lane & 15;
    const int hi    = lane >> 4;          // 0: lanes 0-15, 1: lanes 16-31
    const int khalf = hi * 2;             // K sub-offset (0 or 2) per ISA A/B layout
    const int row   = m0 + l16;           // A row for this lane

    v8f acc[8];
#pragma unroll
    for (int t = 0; t < 8; ++t) acc[t] = (v8f)(0.0f);

    const float* xrow = X + (size_t)row * ldx;

    for (int k = 0; k < HD; k += 4) {
        // A 16x4 tile: lane holds (X[row][k+khalf], X[row][k+khalf+1])
        v2f a = *(const v2f*)(xrow + k + khalf);
#pragma unroll
        for (int t = 0; t < 8; ++t) {
            const int n = t * 16 + l16;
            v2f bm;
            bm.x = W[(k + khalf) * HD + n];
            bm.y = W[(k + khalf + 1) * HD + n];
            acc[t] = __builtin_amdgcn_wmma_f32_16x16x4_f32(
                false, a, false, bm, (short)0, acc[t], false, false);
        }
    }

    // C layout: acc[t][i] -> row m0 + i + 8*hi, col t*16 + l16
#pragma unroll
    for (int i = 0; i < 8; ++i) {
        const int r   = m0 + i + 8 * hi;
        const float di = dinv[r];
        const float di2 = di * di;
        float* hr = h   + (size_t)r * HD;
        float* ar = agg + (size_t)r * HD;
#pragma unroll
        for (int t = 0; t < 8; ++t) {
            const int n = t * 16 + l16;
            const float v = acc[t][i];
            hr[n] = v;
            ar[n] = b[n] + di2 * v;       // bias + self-loop contribution
        }
    }
}

// ---------------- edge aggregation: agg[dst] += dinv[src]*w*dinv[dst] * h[src] ----------------
// One wave per edge; lane handles a float4 chunk (coalesced 512B gather).
__global__ void k_edge(const int* __restrict__ src,
                       const int* __restrict__ dst,
                       const float* __restrict__ w,
                       const float* __restrict__ dinv,
                       const float* __restrict__ h,
                       float* __restrict__ agg) {
    const long long tid = (long long)blockIdx.x * blockDim.x + threadIdx.x;
    const int e    = (int)(tid >> 5);
    const int lane = (int)(tid & 31);
    if (e >= NE) return;

    const int s = src[e];
    const int d = dst[e];
    const float coef = dinv[s] * w[e] * dinv[d];

    const float4 v = reinterpret_cast<const float4*>(h + (size_t)s * HD)[lane];
    float* ap = agg + (size_t)d * HD + lane * 4;
    atomicAdd(ap + 0, coef * v.x);
    atomicAdd(ap + 1, coef * v.y);
    atomicAdd(ap + 2, coef * v.z);
    atomicAdd(ap + 3, coef * v.w);
}

// ---------------- row L2 normalize + ReLU; write into rep (ld = 384) ----------------
__global__ void k_l2relu(const float* __restrict__ agg,
                         float* __restrict__ out /* rep + layer*HD */) {
    const int wave = blockIdx.x * (blockDim.x >> 5) + (threadIdx.x >> 5);
    const int lane = threadIdx.x & 31;
    if (wave >= NN) return;

    float4 v = reinterpret_cast<const float4*>(agg + (size_t)wave * HD)[lane];
    float ss = v.x * v.x + v.y * v.y + v.z * v.z + v.w * v.w;
#pragma unroll
    for (int o = 16; o > 0; o >>= 1) ss += __shfl_xor(ss, o, 32);

    const float nrm = sqrtf(ss);
    const float inv = 1.0f / fmaxf(nrm, EPSV);

    float4 r;
    r.x = fmaxf(v.x * inv, 0.0f);
    r.y = fmaxf(v.y * inv, 0.0f);
    r.z = fmaxf(v.z * inv, 0.0f);
    r.w = fmaxf(v.w * inv, 0.0f);
    // rep row stride is 384 floats; lane chunk offset 16B-aligned
    *reinterpret_cast<float4*>(out + (size_t)wave * LD_REP + lane * 4) = r;
}

// ---------------- final linear: out[N,10] = rep[N,384] @ Wlin[384,10] + blin ----------------
__global__ void k_linear(const float* __restrict__ rep,
                         const float* __restrict__ Wlin,
                         const float* __restrict__ blin,
                         float* __restrict__ out) {
    const int wave = blockIdx.x * (blockDim.x >> 5) + (threadIdx.x >> 5);
    const int lane = threadIdx.x & 31;
    if (wave >= NN) return;

    float acc[NC];
#pragma unroll
    for (int c = 0; c < NC; ++c) acc[c] = 0.0f;

    const float* r = rep + (size_t)wave * LD_REP;
    for (int f = lane; f < LD_REP; f += 32) {
        const float x = r[f];
        const float* wl = Wlin + (size_t)f * NC;
#pragma unroll
        for (int c = 0; c < NC; ++c) acc[c] += x * wl[c];
    }
#pragma unroll
    for (int c = 0; c < NC; ++c) {
#pragma unroll
        for (int o = 16; o > 0; o >>= 1) acc[c] += __shfl_xor(acc[c], o, 32);
    }
    if (lane == 0) {
        float* op = out + (size_t)wave * NC;
#pragma unroll
        for (int c = 0; c < NC; ++c) op[c] = acc[c] + blin[c];
    }
}

// ---------------- launch ----------------
static inline size_t align256(size_t x) { return (x + 255) & ~(size_t)255; }

extern "C" void kernel_launch(void* const* d_in, const int* in_sizes, int n_in,
                              void* d_out, int out_size, void* d_ws, size_t ws_size,
                              hipStream_t stream) {
    const float* x    = (const float*)d_in[0];
    const int*   ei   = (const int*)d_in[1];     // [2,E]: row0 = src, row1 = dst
    const float* ew   = (const float*)d_in[2];
    const float* W1   = (const float*)d_in[3];
    const float* b1   = (const float*)d_in[4];
    const float* W2   = (const float*)d_in[5];
    const float* b2   = (const float*)d_in[6];
    const float* W3   = (const float*)d_in[7];
    const float* b3   = (const float*)d_in[8];
    const float* Wlin = (const float*)d_in[9];
    const float* blin = (const float*)d_in[10];
    float* out = (float*)d_out;

    const int* srcI = ei;
    const int* dstI = ei + NE;

    // workspace carve-up
    char* ws = (char*)d_ws;
    size_t off = 0;
    float* dinv = (float*)(ws + off); off = align256(off + (size_t)NN * 4);
    float* h    = (float*)(ws + off); off = align256(off + (size_t)NN * HD * 4);
    float* agg  = (float*)(ws + off); off = align256(off + (size_t)NN * HD * 4);
    float* rep  = (float*)(ws + off); off = align256(off + (size_t)NN * LD_REP * 4);
    (void)ws_size;

    const int BT = 256;
    const dim3 blk(BT);
    const int gNodes  = (NN + BT - 1) / BT;
    const int gEdges  = (NE + BT - 1) / BT;
    const int gGemm   = ((NN / 16) + (BT / 32) - 1) / (BT / 32);      // 3125 waves
    const int gWaveN  = (NN * 32 + BT - 1) / BT;                      // wave per node
    const int gEdgeW  = (int)(((long long)NE * 32 + BT - 1) / BT);    // wave per edge

    // degrees -> dinv
    k_deg_init<<<gNodes, blk, 0, stream>>>(dinv);
    k_deg_acc<<<gEdges, blk, 0, stream>>>(dstI, ew, dinv);
    k_dinv<<<gNodes, blk, 0, stream>>>(dinv);

    // layer 1: X (ld=128) -> rep[:, 0:128]
    k_gemm_wmma<<<gGemm, blk, 0, stream>>>(x, HD, W1, b1, dinv, h, agg);
    k_edge<<<gEdgeW, blk, 0, stream>>>(srcI, dstI, ew, dinv, h, agg);
    k_l2relu<<<gWaveN, blk, 0, stream>>>(agg, rep + 0);

    // layer 2: rep[:,0:128] (ld=384) -> rep[:, 128:256]
    k_gemm_wmma<<<gGemm, blk, 0, stream>>>(rep + 0, LD_REP, W2, b2, dinv, h, agg);
    k_edge<<<gEdgeW, blk, 0, stream>>>(srcI, dstI, ew, dinv, h, agg);
    k_l2relu<<<gWaveN, blk, 0, stream>>>(agg, rep + HD);

    // layer 3: rep[:,128:256] (ld=384) -> rep[:, 256:384]
    k_gemm_wmma<<<gGemm, blk, 0, stream>>>(rep + HD, LD_REP, W3, b3, dinv, h, agg);
    k_edge<<<gEdgeW, blk, 0, stream>>>(srcI, dstI, ew, dinv, h, agg);
    k_l2relu<<<gWaveN, blk, 0, stream>>>(agg, rep + 2 * HD);

    // final linear
    k_linear<<<gWaveN, blk, 0, stream>>>(rep, Wlin, blin, out);
    (void)in_sizes; (void)n_in; (void)out_size;
}